// Criterion_33947421508242
// MI455X (gfx1250) — compile-verified
//
#include <hip/hip_runtime.h>
#include <stdint.h>

// ---- problem constants (from reference) -----------------------------------
#define B_SZ 2048
#define N_AG 8
#define K_M  6
#define T_ST 60
#define N_LV 3                           // N_levels + 1
#define LOG_2PI_E 2.8378770664093453f    // 1 + ln(2*pi)
#define LOG4      1.3862943611198906f    // ln(4)
#define ENT_W 40.0f
#define KL_W  20.0f

#define WPB    4                         // waves per block
#define BLOCKS 2048
#define NUM_WAVES (BLOCKS * WPB)         // 8192
#define TASKS  (N_LV * B_SZ * N_AG)      // 49152
#define TPW    (TASKS / NUM_WAVES)       // 6 tasks per wave (levels 0..2 x 2 rows)
#define SLICE  1824                      // K*T*5 = 1800 f32, padded

typedef __attribute__((ext_vector_type(4))) unsigned int v4u;
typedef __attribute__((ext_vector_type(4))) int          v4i;
typedef __attribute__((ext_vector_type(8))) int          v8i;

__device__ __forceinline__ float wave_sum32(float v) {
#pragma unroll
  for (int off = 16; off > 0; off >>= 1)
    v += __shfl_xor(v, off, 32);
  return v;
}

// ---- task -> pointer helpers ----------------------------------------------
__device__ __forceinline__ const float* trj_ptr(int task, const float* t0,
                                                const float* t1, const float* t2) {
  const int l = task / (B_SZ * N_AG);
  const int r = task - l * (B_SZ * N_AG);
  const float* tr = (l == 0) ? t0 : (l == 1) ? t1 : t2;
  return tr + (size_t)r * (K_M * T_ST * 5);
}
__device__ __forceinline__ const float* prb_ptr(int task, const float* p0,
                                                const float* p1, const float* p2) {
  const int l = task / (B_SZ * N_AG);
  const int r = task - l * (B_SZ * N_AG);
  const float* pr = (l == 0) ? p0 : (l == 1) ? p1 : p2;
  return pr + (size_t)r * K_M;
}

// ---- TDM: async-stream one 7200B trajectory tile into LDS -----------------
__device__ __forceinline__ void tdm_issue(const float* src, unsigned lds_off) {
  const uint64_t ga = (uint64_t)(uintptr_t)src;
  v4u g0;
  g0.x = 1u;                                    // count=1 (valid user D#)
  g0.y = lds_off;                               // lds_addr (bytes)
  g0.z = (unsigned)(ga & 0xffffffffu);          // global_addr[31:0]
  g0.w = (unsigned)((ga >> 32) & 0x01ffffffu)   // global_addr[56:32]
       | (2u << 30);                            // type = 2 ("image")
  v8i g1;
  g1[0] = (int)(2u << 16);       // workgroup_mask=0, data_size=2 (4 bytes)
  g1[1] = (int)(1800u << 16);    // tensor_dim0[15:0]
  g1[2] = (int)(1u << 16);       // tensor_dim0 hi=0 | tensor_dim1=1
  g1[3] = (int)(1800u << 16);    // tensor_dim1 hi=0 | tile_dim0=1800
  g1[4] = 1;                     // tile_dim1=1, tile_dim2=0
  g1[5] = 1800;                  // tensor_dim0_stride[31:0]
  g1[6] = (int)(1800u << 16);    // stride0 hi=0 | tensor_dim1_stride lo
  g1[7] = 0;
  v4i gz = {0, 0, 0, 0};
#if defined(__clang_major__) && __clang_major__ >= 23
  v8i gz8 = {0, 0, 0, 0, 0, 0, 0, 0};
  __builtin_amdgcn_tensor_load_to_lds(g0, g1, gz, gz, gz8, 0);
#else
  __builtin_amdgcn_tensor_load_to_lds(g0, g1, gz, gz, 0);
#endif
}

// ---- per-timestep accumulation (2 fast logs, rcp, fast sqrt) --------------
__device__ __forceinline__ void accum_t(const float* rec, float x, float y, float m,
                                        float& nacc, float& eacc, float& aacc,
                                        float& dm) {
  const float mux = rec[0], muy = rec[1];
  const float sx  = rec[2], sy  = rec[3], rho = rec[4];
  const float dx = x - mux, dy = y - muy;
  const float rsx = __builtin_amdgcn_rcpf(sx);
  const float rsy = __builtin_amdgcn_rcpf(sy);
  const float lsxsy = __logf(sx * sy);
  nacc += (LOG4 + lsxsy + fabsf(dx) * rsx + fabsf(dy) * rsy) * m;   // Laplace NLL
  eacc += LOG_2PI_E + lsxsy + 0.5f * __logf((1.f - rho) * (1.f + rho));
  const float d = __builtin_amdgcn_sqrtf(dx * dx + dy * dy);
  aacc += d * m;
  dm = d * m;
}

__global__ __launch_bounds__(WPB * 32)
void criterion_kernel(const float* __restrict__ tr0, const float* __restrict__ pr0,
                      const float* __restrict__ tr1, const float* __restrict__ pr1,
                      const float* __restrict__ tr2, const float* __restrict__ pr2,
                      const float* __restrict__ gt,  float* __restrict__ partial)
{
  __shared__ float lds[WPB * 2 * SLICE];
  __shared__ float sred[WPB];

  const int lane = (int)(threadIdx.x & 31u);
  const int wib  = __builtin_amdgcn_readfirstlane((int)(threadIdx.x >> 5));
  const int wave = (int)blockIdx.x * WPB + wib;

  float* buf0 = &lds[wib * 2 * SLICE];
  const unsigned off0 = (unsigned)(uintptr_t)(void*)buf0;   // LDS byte offset
  const unsigned off1 = off0 + SLICE * 4u;

  // prologue: fill both pipeline stages
  tdm_issue(trj_ptr(wave,             tr0, tr1, tr2), off0);
  tdm_issue(trj_ptr(wave + NUM_WAVES, tr0, tr1, tr2), off1);

  const bool has2 = lane < (T_ST - 32);                     // lanes 0..27 own t+32

  // gt rows: task r-index alternates between `wave` (even i) and
  // `wave + NUM_WAVES` (odd i) -- the 3 levels reuse the same two rows.
  float gx0[2], gy0[2], gm0[2], gx1[2], gy1[2], gm1[2];
#pragma unroll
  for (int p = 0; p < 2; ++p) {
    const float* gtb = gt + (size_t)(wave + p * NUM_WAVES) * (T_ST * 3);
    gx0[p] = gtb[lane * 3 + 0]; gy0[p] = gtb[lane * 3 + 1]; gm0[p] = gtb[lane * 3 + 2];
    gx1[p] = 0.f; gy1[p] = 0.f; gm1[p] = 0.f;
    if (has2) {
      const int o = (lane + 32) * 3;
      gx1[p] = gtb[o + 0]; gy1[p] = gtb[o + 1]; gm1[p] = gtb[o + 2];
    }
  }

  const float scale = (float)N_AG / ((float)N_LV * (float)B_SZ);
  float acc = 0.f;

#pragma unroll 1                          // keep the hot loop I$-resident
  for (int i = 0; i < TPW; ++i) {
    const int task = wave + i * NUM_WAVES;
    const int p = i & 1;
    const float x0 = gx0[p], y0 = gy0[p], m0 = gm0[p];
    const float x1 = gx1[p], y1 = gy1[p], m1 = gm1[p];

    // priors for this (level, b, n) -- overlaps the in-flight TDM
    const float* prb = prb_ptr(task, pr0, pr1, pr2);
    float pk[K_M];
#pragma unroll
    for (int k = 0; k < K_M; ++k) pk[k] = prb[k];

    // steady state keeps one TDM in flight; drain fully on the last task
    if (i == TPW - 1) __builtin_amdgcn_s_wait_tensorcnt(0);
    else              __builtin_amdgcn_s_wait_tensorcnt(1);
    asm volatile("" ::: "memory");

    const float* slice = buf0 + p * SLICE;
    float nll[K_M], ent[K_M], ade[K_M], fde[K_M];
#pragma unroll
    for (int k = 0; k < K_M; ++k) {
      const float* rk = slice + k * (T_ST * 5);
      float nacc = 0.f, eacc = 0.f, aacc = 0.f, dm, facc = 0.f;
      accum_t(rk + lane * 5, x0, y0, m0, nacc, eacc, aacc, dm);
      if (has2) {
        accum_t(rk + (lane + 32) * 5, x1, y1, m1, nacc, eacc, aacc, dm);
        if (lane == (T_ST - 1 - 32)) facc = dm;             // FDE lives on lane 27
      }
      nll[k] = wave_sum32(nacc);                            // masked Σ_t NLL
      ent[k] = wave_sum32(eacc);                            // unmasked Σ_t entropy
      ade[k] = wave_sum32(aacc) * (1.0f / (float)T_ST);     // mean_t
      fde[k] = __shfl(facc, T_ST - 1 - 32, 32);             // broadcast, no butterfly
    }

    // K=6 posterior softmax / KL / entropy-max / min(FDE+ADE)
    float lpu[K_M], mx = -1e30f;
#pragma unroll
    for (int k = 0; k < K_M; ++k) { lpu[k] = -nll[k] + __logf(pk[k]); mx = fmaxf(mx, lpu[k]); }
    float ex[K_M], z = 0.f;
#pragma unroll
    for (int k = 0; k < K_M; ++k) { ex[k] = __expf(lpu[k] - mx); z += ex[k]; }
    const float invz = __builtin_amdgcn_rcpf(z), logz = __logf(z);
    float loss1 = 0.f, kl = 0.f, entmax = -1e30f, famin = 1e30f;
#pragma unroll
    for (int k = 0; k < K_M; ++k) {
      const float post = ex[k] * invz;
      loss1 += nll[k] * post;
      kl    += post * (-nll[k] - mx - logz);                // post*(log post - log p)
      entmax = fmaxf(entmax, ent[k]);
      famin  = fminf(famin, fde[k] + ade[k]);
    }
    acc += (loss1 + ENT_W * entmax + KL_W * kl + 100.0f * famin) * scale;

    // refill the buffer we just drained
    if (i < TPW - 2) {
      asm volatile("s_wait_dscnt 0x0" ::: "memory");        // LDS reads drained
      tdm_issue(trj_ptr(wave + (i + 2) * NUM_WAVES, tr0, tr1, tr2),
                p ? off1 : off0);
    }
  }

  // deterministic block reduction -> one partial per block
  if (lane == 0) sred[wib] = acc;
  __syncthreads();
  if (threadIdx.x == 0) {
    float s = 0.f;
#pragma unroll
    for (int j = 0; j < WPB; ++j) s += sred[j];
    partial[blockIdx.x] = s;
  }
}

// ---- stage 2: deterministic reduction of 2048 block partials --------------
__global__ __launch_bounds__(256)
void reduce_kernel(const float* __restrict__ partial, float* __restrict__ out) {
  __shared__ float s[8];
  const int tid = (int)threadIdx.x;
  float v = 0.f;
  for (int j = tid; j < BLOCKS; j += 256) v += partial[j];
  v = wave_sum32(v);
  if ((tid & 31) == 0) s[tid >> 5] = v;
  __syncthreads();
  if (tid == 0) {
    float t = 0.f;
#pragma unroll
    for (int j = 0; j < 8; ++j) t += s[j];
    out[0] = t;
  }
}

extern "C" void kernel_launch(void* const* d_in, const int* in_sizes, int n_in,
                              void* d_out, int out_size, void* d_ws, size_t ws_size,
                              hipStream_t stream) {
  (void)in_sizes; (void)n_in; (void)out_size; (void)ws_size;
  const float* tr0 = (const float*)d_in[0];
  const float* pr0 = (const float*)d_in[1];
  const float* tr1 = (const float*)d_in[2];
  const float* pr1 = (const float*)d_in[3];
  const float* tr2 = (const float*)d_in[4];
  const float* pr2 = (const float*)d_in[5];
  const float* gt  = (const float*)d_in[6];
  float* partial = (float*)d_ws;                 // 2048 f32 = 8 KB scratch
  float* out     = (float*)d_out;

  criterion_kernel<<<BLOCKS, WPB * 32, 0, stream>>>(tr0, pr0, tr1, pr1, tr2, pr2,
                                                    gt, partial);
  reduce_kernel<<<1, 256, 0, stream>>>(partial, out);
}